// kernel_generated_1_62904091017356
// MI455X (gfx1250) — compile-verified
//
#include <hip/hip_runtime.h>

typedef float v2f __attribute__((ext_vector_type(2)));
typedef float v8f __attribute__((ext_vector_type(8)));

constexpr int NL  = 128;  // batch
constexpr int NJ  = 12;   // mid channels (stage B input)
constexpr int NI  = 21;   // stage B output channels
constexpr int NO  = 96;   // final channels
constexpr int HW  = 56;   // spatial

// Stage B GEMM: K = 84 (j*7+k), K-stride 84 floats (336 B: 8B aligned, banks %64 = 20)
constexpr int KB   = 84;
constexpr int SB_A = 84;
// Stage C GEMM: K = 63 -> 64 steps, K-stride 68 floats (272 B: 8B aligned, banks %64 = 4)
constexpr int SC   = 68;

// t5 is stored transposed: t5[l][n][i][m]  (m contiguous -> coalesced b128 stores)

// -------------------------------------------------------------------------
// Kernel 0: t3r[l,j,n,m] = x[l,j,n,(m-1)%56] + x[l,12+j,n,(m-1)%56]
// -------------------------------------------------------------------------
__global__ void build_t3r(const float* __restrict__ x, float* __restrict__ t3r) {
    int idx = blockIdx.x * blockDim.x + threadIdx.x;
    const int total = NL * NJ * HW * HW;
    if (idx >= total) return;
    int m = idx % HW;
    int n = (idx / HW) % HW;
    int j = (idx / (HW * HW)) % NJ;
    int l = idx / (HW * HW * NJ);
    int ms = (m + HW - 1) % HW;                      // roll by +1
    size_t b0 = (((size_t)l * 24 + j) * HW + n) * HW + ms;
    size_t b1 = (((size_t)l * 24 + 12 + j) * HW + n) * HW + ms;
    t3r[idx] = x[b0] + x[b1];
}

// -------------------------------------------------------------------------
// Stage B: t5[l,n,i,m] = sum_{j,k} t3r[l,j,n,m+k-3] * w1[j,k,i]
// Per (l,n): M = m (4 tiles of 16), N = i (2 tiles), K = 84 = 21 WMMA steps.
// -------------------------------------------------------------------------
__global__ void __launch_bounds__(256) stageB(const float* __restrict__ t3r,
                                              const float* __restrict__ w1,
                                              float* __restrict__ t5) {
    const int n = blockIdx.x;   // 0..55
    const int l = blockIdx.y;   // 0..127
    __shared__ float sT[NJ * HW];       // raw input slab [j][m]    672 f
    __shared__ float sA[64 * SB_A];     // windowed A     [m][kk]   5376 f
    __shared__ float sW[32 * SB_A];     // weights        [i][kk]   2688 f
    const int tid = threadIdx.x;

    for (int idx = tid; idx < NJ * HW; idx += 256) {
        int j = idx / HW, m = idx % HW;
        sT[idx] = t3r[(((size_t)l * NJ + j) * HW + n) * HW + m];
    }
    for (int idx = tid; idx < 32 * KB; idx += 256) {
        int i = idx / KB, kk = idx - i * KB;          // kk = j*7+k
        sW[i * SB_A + kk] = (i < NI) ? w1[kk * NI + i] : 0.0f;
    }
    __syncthreads();
    // Pre-window A: all shifts/padding resolved once, outside the K loop.
    for (int idx = tid; idx < 64 * KB; idx += 256) {
        int m = idx / KB, kk = idx - (idx / KB) * KB;
        int jj = kk / 7, k = kk - 7 * jj;
        int mm = m + k - 3;
        float v = 0.0f;
        if (m < HW && mm >= 0 && mm < HW) v = sT[jj * HW + mm];
        sA[m * SB_A + kk] = v;
    }
    __syncthreads();

    const int wave   = tid >> 5;
    const int lane   = tid & 31;
    const int lane16 = lane & 15;
    const int khalf  = (lane >> 4) << 1;             // 0 or 2

    const int mt = wave & 3, nt = wave >> 2;         // 8 waves <-> 8 tiles
    const int b0 = mt * 16, i0 = nt * 16;
    const int aBase = (b0 + lane16) * SB_A + khalf;
    const int bBase = (i0 + lane16) * SB_A + khalf;

    v8f c = {};
    #pragma unroll
    for (int s = 0; s < 21; ++s) {                   // K = 84 = 21 * 4
        v2f av = *(const v2f*)&sA[aBase + 4 * s];    // ds_load_b64 (paired by compiler)
        v2f bv = *(const v2f*)&sW[bBase + 4 * s];
        c = __builtin_amdgcn_wmma_f32_16x16x4_f32(false, av, false, bv,
                                                  (short)0, c, false, false);
    }

    // D rows contiguous along m with t5[l][n][i][m] -> two b128 stores per lane
    const int icol  = i0 + lane16;
    const int mbase = b0 + ((lane >> 4) << 3);
    if (icol < NI && mbase + 7 < HW) {
        float* dst = t5 + (((size_t)l * HW + n) * NI + icol) * HW + mbase;
        *(float4*)(dst)     = make_float4(c[0], c[1], c[2], c[3]);
        *(float4*)(dst + 4) = make_float4(c[4], c[5], c[6], c[7]);
    }
}

// -------------------------------------------------------------------------
// Stage C: y[l,j,a,b] = sum_{k,i} t5[l,a+k-1,i,b] * w0[j,k,i]
// Per (l,a): M = b (4 tiles), N = j (6 tiles), K = 63 pad 64 = 16 WMMA steps.
// -------------------------------------------------------------------------
__global__ void __launch_bounds__(256) stageC(const float* __restrict__ t5,
                                              const float* __restrict__ w0,
                                              float* __restrict__ y) {
    const int a = blockIdx.x;   // output height
    const int l = blockIdx.y;
    __shared__ float sA[64 * SC];   // [b][kk]  4352 f
    __shared__ float sB[NO * SC];   // [j][kk]  6528 f
    const int tid = threadIdx.x;

    // Coalesced A fill: fast axis = b matches t5's contiguous m axis.
    for (int idx = tid; idx < SC * 64; idx += 256) {
        int b = idx & 63, kk = idx >> 6;              // kk = k*21+i (0..67)
        float v = 0.0f;
        if (b < HW && kk < 63) {
            int k = kk / NI, i = kk - k * NI;
            int nrow = a + k - 1;
            if (nrow >= 0 && nrow < HW)
                v = t5[(((size_t)l * HW + nrow) * NI + i) * HW + b];
        }
        sA[b * SC + kk] = v;
    }
    for (int idx = tid; idx < NO * SC; idx += 256) {
        int j = idx / SC, kk = idx - j * SC;
        float v = 0.0f;
        if (kk < 63) {
            int k = kk / NI, i = kk - k * NI;
            v = w0[((size_t)j * 3 + k) * NI + i];
        }
        sB[idx] = v;
    }
    __syncthreads();

    const int wave   = tid >> 5;
    const int lane   = tid & 31;
    const int lane16 = lane & 15;
    const int khalf  = (lane >> 4) << 1;

    for (int t = wave; t < 24; t += 8) {             // 4 Mtiles x 6 Ntiles
        const int mt = t & 3, nt = t >> 2;
        const int b0 = mt * 16, j0 = nt * 16;
        const int aBase = (b0 + lane16) * SC + khalf;
        const int bBase = (j0 + lane16) * SC + khalf;

        v8f c = {};
        #pragma unroll
        for (int s = 0; s < 16; ++s) {               // K = 64 (63 + 1 zero)
            v2f av = *(const v2f*)&sA[aBase + 4 * s];
            v2f bv = *(const v2f*)&sB[bBase + 4 * s];
            c = __builtin_amdgcn_wmma_f32_16x16x4_f32(false, av, false, bv,
                                                      (short)0, c, false, false);
        }

        const int jcol  = j0 + lane16;
        const int mbase = b0 + ((lane >> 4) << 3);
        if (mbase + 7 < HW) {
            float* dst = y + (((size_t)l * NO + jcol) * HW + a) * HW + mbase;
            *(float4*)(dst)     = make_float4(c[0], c[1], c[2], c[3]);
            *(float4*)(dst + 4) = make_float4(c[4], c[5], c[6], c[7]);
        }
    }
}

// -------------------------------------------------------------------------
extern "C" void kernel_launch(void* const* d_in, const int* in_sizes, int n_in,
                              void* d_out, int out_size, void* d_ws, size_t ws_size,
                              hipStream_t stream) {
    const float* x  = (const float*)d_in[0];   // (128,24,56,56)
    const float* w0 = (const float*)d_in[1];   // (96,3,21)
    const float* w1 = (const float*)d_in[2];   // (12,7,21)
    float* y = (float*)d_out;                  // (128,96,56,56)

    float* t3r = (float*)d_ws;                               // 19.3 MB
    float* t5  = t3r + (size_t)NL * NJ * HW * HW;            // 33.7 MB, [l][n][i][m]

    {
        int total = NL * NJ * HW * HW;
        build_t3r<<<(total + 255) / 256, 256, 0, stream>>>(x, t3r);
    }
    dim3 grid(HW, NL);
    stageB<<<grid, 256, 0, stream>>>(t3r, w1, t5);
    stageC<<<grid, 256, 0, stream>>>(t5, w0, y);
}